// TopKAbsolutes2D_43800076484737
// MI455X (gfx1250) — compile-verified
//
#include <hip/hip_runtime.h>
#include <stdint.h>

// ---------------------------------------------------------------------------
// Per-row top-k-|x| sparsification (B=128 rows, N=262144 fp32, k=2048).
// One 32-wave workgroup per row. 3-level radix select on the 31-bit |x|
// bit pattern (11/11/9 bits), with:
//   * global->LDS streaming via CDNA5 GLOBAL_LOAD_ASYNC_TO_LDS_B128
//     (ASYNCcnt), 32 KB tiles, triple-buffered => 1 barrier per tile
//   * parallel hierarchical bin scan (64-way partials, no 2048-long serial walk)
//   * pass-2 runs from an LDS-compacted candidate list (expected ~1.7K keys,
//     8192 cap) -- removes one full 134 MB re-read; streaming fallback if full
// Traffic: 1 HBM read + 2 L2 re-reads (+ compaction) + 1 HBM write
// => near the ~11.5 us roofline at 23.3 TB/s (all 134 MB is L2-resident).
// ---------------------------------------------------------------------------

#define TPB      1024                 // 32 waves (wave32)
#define PER_LANE 8                    // 8 floats = 32 B per lane per tile
#define CHUNK    (TPB * PER_LANE)     // 8192 floats = 32 KB per tile
#define NBUF     3                    // triple buffer
#define CANDMAX  8192                 // LDS candidate capacity (keys)

// copies 16 bytes from global address sbase + voff to LDS address ldsOff   (GVS mode, async, ASYNCcnt)
__device__ __forceinline__ void async_copy16(uint32_t ldsOff, uint64_t sbase,
                                             uint32_t voff) {
  asm volatile("global_load_async_to_lds_b128 %0, %1, %2"
               :: "v"(ldsOff), "v"(voff), "s"(sbase)
               : "memory");
}

// Wait for this wave's outstanding async-LDS ops: tile in flight issues 2 ops.
__device__ __forceinline__ void wait_async(bool next_in_flight) {
  if (next_in_flight) asm volatile("s_wait_asynccnt 0x2" ::: "memory");
  else                asm volatile("s_wait_asynccnt 0x0" ::: "memory");
}

__global__ void __launch_bounds__(TPB)
topk_abs_rows(const float* __restrict__ x, const int* __restrict__ topk_p,
              float* __restrict__ out, int N) {
  __shared__ __align__(16) uint32_t stage[NBUF][CHUNK];  // 96 KB
  __shared__ unsigned int hist[2048];                    //  8 KB
  __shared__ unsigned int cand[CANDMAX];                 // 32 KB
  __shared__ unsigned int part[64];
  __shared__ unsigned int s_bin, s_target, s_cbin, s_cnt, s_take;

  const int tid = threadIdx.x;
  const int row = blockIdx.x;
  const uint64_t sbase = (uint64_t)(uintptr_t)(x + (size_t)row * (size_t)N);
  float* rowOut = out + (size_t)row * (size_t)N;
  const unsigned int K = (unsigned int)*topk_p;
  const int nchunk = N / CHUNK;                 // 262144/8192 = 32 tiles
  const uint32_t myByte = (uint32_t)(tid * 32); // 32 B per lane per tile

  uint32_t ldsBase[NBUF];
#pragma unroll
  for (int b = 0; b < NBUF; ++b)
    ldsBase[b] = (uint32_t)(uintptr_t)&stage[b][tid * PER_LANE];

  // ---- triple-buffered async streaming over the row: body(curPtr, tileIdx) ----
  auto stream = [&](auto&& body) {
    async_copy16(ldsBase[0],      sbase, myByte);
    async_copy16(ldsBase[0] + 16, sbase, myByte + 16);
    for (int c = 0; c < nchunk; ++c) {
      const bool more = (c + 1) < nchunk;
      if (more) {
        const uint32_t dst = ldsBase[(c + 1) % NBUF];
        const uint32_t go  = (uint32_t)((c + 1) * CHUNK * 4) + myByte;
        async_copy16(dst,      sbase, go);
        async_copy16(dst + 16, sbase, go + 16);
      }
      wait_async(more);          // my tile-c pair landed (in-order completion)
      __syncthreads();           // all waves' tile-c data visible; also fences
                                 // reads of the buffer being refilled next iter
      body(&stage[c % NBUF][tid * PER_LANE], c);
    }
  };

  // ---- parallel suffix scan over `nb` bins: find bin holding rank `target`
  // (counting from the top). Leading barrier fences the last tile's atomics.
  auto select_bin = [&](unsigned nb, unsigned target) {
    __syncthreads();
    const unsigned ng = nb / 32;               // 64 or 16 groups
    if (tid < (int)ng) {
      unsigned acc = 0;
      const unsigned hi = nb - 1 - 32 * (unsigned)tid;  // descending bins
#pragma unroll
      for (int i = 0; i < 32; ++i) acc += hist[hi - i];
      part[tid] = acc;
    }
    __syncthreads();
    if (tid == 0) {
      unsigned acc = 0, g = 0;
      for (; g < ng - 1; ++g) {                // <=63 serial steps
        unsigned h = part[g];
        if (acc + h >= target) break;
        acc += h;
      }
      unsigned b  = nb - 1 - 32 * g;           // walk <=32 bins in group g
      const unsigned lo = b - 31;
      for (;;) {
        unsigned h = hist[b];
        if (acc + h >= target || b == lo) break;
        acc += h;
        --b;
      }
      s_bin    = b;
      s_target = target - acc;                 // remaining rank inside bin b
      s_cbin   = hist[b];                      // bin population (for compaction)
    }
    __syncthreads();
  };

  // ================= pass 0: bins = key >> 20 (11 bits) =================
  for (int i = tid; i < 2048; i += TPB) hist[i] = 0u;
  __syncthreads();
  stream([&](const uint32_t* cur, int) {
#pragma unroll
    for (int j = 0; j < PER_LANE; ++j) {
      unsigned k = cur[j] & 0x7FFFFFFFu;
      atomicAdd(&hist[k >> 20], 1u);
    }
  });
  select_bin(2048u, K);
  const unsigned b1 = s_bin;
  unsigned target   = s_target;
  const unsigned C  = s_cbin;                  // population of bin b1
  const bool useLds = (C <= CANDMAX);

  // ====== pass 1: within bin b1, bins = (key >> 9) & 0x7FF; compact keys ======
  for (int i = tid; i < 2048; i += TPB) hist[i] = 0u;
  if (tid == 0) s_cnt = 0u;
  __syncthreads();
  stream([&](const uint32_t* cur, int) {
#pragma unroll
    for (int j = 0; j < PER_LANE; ++j) {
      unsigned k = cur[j] & 0x7FFFFFFFu;
      if ((k >> 20) == b1) {
        atomicAdd(&hist[(k >> 9) & 0x7FFu], 1u);
        if (useLds) {
          unsigned i = atomicAdd(&s_cnt, 1u);
          if (i < CANDMAX) cand[i] = k;
        }
      }
    }
  });
  select_bin(2048u, target);
  const unsigned prefix22 = (b1 << 11) | s_bin;
  target = s_target;

  // ====== pass 2: within prefix22, bins = key & 0x1FF (9 bits) ======
  for (int i = tid; i < 512; i += TPB) hist[i] = 0u;
  __syncthreads();
  if (useLds) {
    for (unsigned i = tid; i < C; i += TPB) {  // entirely from LDS candidates
      unsigned k = cand[i];
      if ((k >> 9) == prefix22) atomicAdd(&hist[k & 0x1FFu], 1u);
    }
  } else {                                     // fallback: stream the row again
    stream([&](const uint32_t* cur, int) {
#pragma unroll
      for (int j = 0; j < PER_LANE; ++j) {
        unsigned k = cur[j] & 0x7FFFFFFFu;
        if ((k >> 9) == prefix22) atomicAdd(&hist[k & 0x1FFu], 1u);
      }
    });
  }
  select_bin(512u, target);
  const unsigned T   = (prefix22 << 9) | s_bin;  // exact 31-bit threshold key
  const unsigned nEq = s_target;                 // #(==T) elements to keep
  if (tid == 0) s_take = 0u;
  __syncthreads();

  // ====================== write pass ======================
  stream([&](const uint32_t* cur, int c) {
    float4 o0, o1;
    float* op = (float*)&o0;  // o0 then o1 contiguous in j below
#pragma unroll
    for (int j = 0; j < PER_LANE; ++j) {
      if (j == 4) op = (float*)&o1;
      unsigned u = cur[j];
      unsigned k = u & 0x7FFFFFFFu;
      bool keep = (k > T);
      if (k == T) keep = (atomicAdd(&s_take, 1u) < nEq);  // cap ties at exactly k
      op[j & 3] = keep ? __uint_as_float(u) : 0.0f;
    }
    float* dst = rowOut + (size_t)c * CHUNK + tid * PER_LANE;
    *reinterpret_cast<float4*>(dst)     = o0;
    *reinterpret_cast<float4*>(dst + 4) = o1;
  });
}

extern "C" void kernel_launch(void* const* d_in, const int* in_sizes, int n_in,
                              void* d_out, int out_size, void* d_ws, size_t ws_size,
                              hipStream_t stream) {
  const float* x      = (const float*)d_in[0];
  const int*   topk_p = (const int*)d_in[1];
  float*       out    = (float*)d_out;

  const int B = 128;               // x.shape[0] per reference
  const int N = in_sizes[0] / B;   // 262144 per row (multiple of CHUNK)

  topk_abs_rows<<<dim3(B), dim3(TPB), 0, stream>>>(x, topk_p, out, N);
}